// GNN_17832704213427
// MI455X (gfx1250) — compile-verified
//
#include <hip/hip_runtime.h>
#include <hip/hip_bf16.h>

typedef float v8f __attribute__((ext_vector_type(8)));
typedef float v2f __attribute__((ext_vector_type(2)));

#define N_NODES 50000
#define N_EDGES 800000
#define DHID    128

// ---------------------------------------------------------------------------
// GEMM: out = X @ W + b for one of {q,k,v,skip}, selected by blockIdx.y.
// One wave computes a 16x16 f32 tile with V_WMMA_F32_16X16X4_F32 (K-steps of 4).
// q/k/v go to qkv[N,384]; skip goes to agg[N,128] (pre-seeding the aggregator).
// ---------------------------------------------------------------------------
__global__ __launch_bounds__(256)
void gemm_qkvs(const float* __restrict__ X,
               const float* __restrict__ Wq, const float* __restrict__ bq,
               const float* __restrict__ Wk, const float* __restrict__ bk,
               const float* __restrict__ Wv, const float* __restrict__ bv,
               const float* __restrict__ Ws, const float* __restrict__ bs,
               float* __restrict__ qkv, float* __restrict__ agg)
{
    __shared__ float As[16 * 132];                 // 16x128 x-tile, padded stride
    const int rowBase = blockIdx.x * 16;
    const int proj    = blockIdx.y;                // 0=q 1=k 2=v 3=skip
    const int wave    = threadIdx.x >> 5;
    const int lane    = threadIdx.x & 31;

    for (int i = threadIdx.x; i < 16 * 128; i += 256) {
        int r = i >> 7, c = i & 127;
        As[r * 132 + c] = X[(rowBase + r) * 128 + c];
    }
    __syncthreads();

    const float* W; const float* Bb;
    if      (proj == 0) { W = Wq; Bb = bq; }
    else if (proj == 1) { W = Wk; Bb = bk; }
    else if (proj == 2) { W = Wv; Bb = bv; }
    else                { W = Ws; Bb = bs; }

    const int colBase = wave * 16;
    const int hi  = lane >> 4;                     // lane half selects K pair
    const int m16 = lane & 15;

    v8f acc = {};
    #pragma unroll 4
    for (int k = 0; k < 128; k += 4) {
        v2f a, b;
        a.x = As[m16 * 132 + k + 2 * hi];
        a.y = As[m16 * 132 + k + 2 * hi + 1];
        b.x = W[(k + 2 * hi)     * 128 + colBase + m16];
        b.y = W[(k + 2 * hi + 1) * 128 + colBase + m16];
        acc = __builtin_amdgcn_wmma_f32_16x16x4_f32(
                  false, a, false, b, (short)0, acc, false, false);
    }

    const int   col  = colBase + m16;
    const float bias = Bb[col];
    if (proj < 3) {
        float* out = qkv + proj * 128;
        #pragma unroll
        for (int i = 0; i < 8; ++i)
            out[(size_t)(rowBase + i + 8 * hi) * 384 + col] = acc[i] + bias;
    } else {
        #pragma unroll
        for (int i = 0; i < 8; ++i)
            agg[(size_t)(rowBase + i + 8 * hi) * 128 + col] = acc[i] + bias;
    }
}

// ---------------------------------------------------------------------------
// Init per-layer segment buffers: m_enc=0 (monotone encoding of -inf floor), den=0
// ---------------------------------------------------------------------------
__global__ void init_md(unsigned* __restrict__ menc, float* __restrict__ den, int n)
{
    int i = blockIdx.x * 256 + threadIdx.x;
    if (i < n) { menc[i] = 0u; den[i] = 0.0f; }
}

__device__ __forceinline__ unsigned f2ord(float f) {
    unsigned b = __float_as_uint(f);
    return (b & 0x80000000u) ? ~b : (b | 0x80000000u);
}
__device__ __forceinline__ float ord2f(unsigned u) {
    return (u & 0x80000000u) ? __uint_as_float(u ^ 0x80000000u)
                             : __uint_as_float(~u);
}

// ---------------------------------------------------------------------------
// Pass 1: per-edge logits (8 heads, HEAD_DIM=16) + segment max via atomicMax
// on monotone-uint encoding. One wave per edge; lane loads float4 (32*4=128).
// Head h = lanes 4h..4h+3; 4-lane shfl_xor reduction.
// ---------------------------------------------------------------------------
__global__ __launch_bounds__(256)
void edge_logits(const float* __restrict__ qkv,
                 const int* __restrict__ ei, const int* __restrict__ et,
                 const float* __restrict__ Eemb,     // [4,128] this layer
                 float* __restrict__ exbuf,          // [E,8]
                 unsigned* __restrict__ menc)        // [N,8]
{
    const int wave = threadIdx.x >> 5, lane = threadIdx.x & 31;
    const int e = blockIdx.x * 8 + wave;
    const int src = ei[e], dst = ei[N_EDGES + e], t = et[e];

    const float4 qv = ((const float4*)(qkv + (size_t)dst * 384))[lane];
    const float4 kv = ((const float4*)(qkv + (size_t)src * 384 + 128))[lane];
    const float4 ev = ((const float4*)(Eemb + t * 128))[lane];

    float p = qv.x * (kv.x + ev.x) + qv.y * (kv.y + ev.y)
            + qv.z * (kv.z + ev.z) + qv.w * (kv.w + ev.w);
    p += __shfl_xor(p, 1, 32);
    p += __shfl_xor(p, 2, 32);

    if ((lane & 3) == 0) {
        int h = lane >> 2;
        float logit = p * 0.25f;                    // 1/sqrt(16)
        exbuf[(size_t)e * 8 + h] = logit;
        atomicMax(&menc[(size_t)dst * 8 + h], f2ord(logit));
    }
}

// ---------------------------------------------------------------------------
// Pass 2: ex = exp(logit - m[dst]); den[dst] += ex.  Thread per (edge, head).
// ---------------------------------------------------------------------------
__global__ void edge_exp(float* __restrict__ exbuf,
                         const int* __restrict__ ei,
                         const unsigned* __restrict__ menc,
                         float* __restrict__ den)
{
    int idx = blockIdx.x * 256 + threadIdx.x;
    if (idx >= N_EDGES * 8) return;
    int e = idx >> 3, h = idx & 7;
    int dst = ei[N_EDGES + e];
    float m  = ord2f(menc[(size_t)dst * 8 + h]);
    float ex = expf(exbuf[idx] - m);
    exbuf[idx] = ex;
    atomicAdd(&den[(size_t)dst * 8 + h], ex);
}

// ---------------------------------------------------------------------------
// Pass 3: agg[dst] += alpha * (v[src] + e).  Wave per edge, 4 f32 atomics/lane.
// agg was pre-seeded with the skip projection (+ bias) by gemm_qkvs.
// ---------------------------------------------------------------------------
__global__ __launch_bounds__(256)
void edge_scatter(const float* __restrict__ qkv,
                  const int* __restrict__ ei, const int* __restrict__ et,
                  const float* __restrict__ Eemb,
                  const float* __restrict__ exbuf,
                  const float* __restrict__ den,
                  float* __restrict__ agg)
{
    const int wave = threadIdx.x >> 5, lane = threadIdx.x & 31;
    const int e = blockIdx.x * 8 + wave;
    const int src = ei[e], dst = ei[N_EDGES + e], t = et[e];

    const float4 vv = ((const float4*)(qkv + (size_t)src * 384 + 256))[lane];
    const float4 ev = ((const float4*)(Eemb + t * 128))[lane];
    const int h = lane >> 2;
    const float alpha = exbuf[(size_t)e * 8 + h] / den[(size_t)dst * 8 + h];

    float* a = agg + (size_t)dst * 128 + lane * 4;
    atomicAdd(a + 0, alpha * (vv.x + ev.x));
    atomicAdd(a + 1, alpha * (vv.y + ev.y));
    atomicAdd(a + 2, alpha * (vv.z + ev.z));
    atomicAdd(a + 3, alpha * (vv.w + ev.w));
}

// ---------------------------------------------------------------------------
// LayerNorm + ReLU: one wave per node, 32-lane reductions.
// ---------------------------------------------------------------------------
__global__ __launch_bounds__(256)
void ln_relu(const float* __restrict__ agg,
             const float* __restrict__ gamma, const float* __restrict__ beta,
             float* __restrict__ out)
{
    const int wave = threadIdx.x >> 5, lane = threadIdx.x & 31;
    const int n = blockIdx.x * 8 + wave;

    float4 v = ((const float4*)(agg + (size_t)n * 128))[lane];
    float s = v.x + v.y + v.z + v.w;
    #pragma unroll
    for (int o = 1; o < 32; o <<= 1) s += __shfl_xor(s, o, 32);
    float mu = s * (1.0f / 128.0f);

    float dx = v.x - mu, dy = v.y - mu, dz = v.z - mu, dw = v.w - mu;
    float q = dx * dx + dy * dy + dz * dz + dw * dw;
    #pragma unroll
    for (int o = 1; o < 32; o <<= 1) q += __shfl_xor(q, o, 32);
    float inv = rsqrtf(q * (1.0f / 128.0f) + 1e-5f);

    float4 g  = ((const float4*)gamma)[lane];
    float4 bb = ((const float4*)beta)[lane];
    float4 y;
    y.x = fmaxf(dx * inv * g.x + bb.x, 0.0f);
    y.y = fmaxf(dy * inv * g.y + bb.y, 0.0f);
    y.z = fmaxf(dz * inv * g.z + bb.z, 0.0f);
    y.w = fmaxf(dw * inv * g.w + bb.w, 0.0f);
    ((float4*)(out + (size_t)n * 128))[lane] = y;
}

// ---------------------------------------------------------------------------
extern "C" void kernel_launch(void* const* d_in, const int* in_sizes, int n_in,
                              void* d_out, int out_size, void* d_ws, size_t ws_size,
                              hipStream_t stream)
{
    const float* x     = (const float*)d_in[0];
    const int*   ei    = (const int*)  d_in[1];   // [2,E]
    const int*   et    = (const int*)  d_in[2];
    const float* Wq    = (const float*)d_in[3];
    const float* bq    = (const float*)d_in[4];
    const float* Wk    = (const float*)d_in[5];
    const float* bk    = (const float*)d_in[6];
    const float* Wv    = (const float*)d_in[7];
    const float* bv    = (const float*)d_in[8];
    const float* Wsk   = (const float*)d_in[9];
    const float* bsk   = (const float*)d_in[10];
    const float* Eemb  = (const float*)d_in[11];  // [2,4,128]
    const float* gamma = (const float*)d_in[12];
    const float* beta  = (const float*)d_in[13];

    // Workspace carve-up (floats)
    float*    ws    = (float*)d_ws;
    float*    qkv   = ws;                              // N*384 = 19.2M
    float*    agg   = qkv   + (size_t)N_NODES * 384;   // N*128 =  6.4M
    float*    h1    = agg   + (size_t)N_NODES * 128;   // N*128 =  6.4M
    float*    exbuf = h1    + (size_t)N_NODES * 128;   // E*8   =  6.4M
    float*    den   = exbuf + (size_t)N_EDGES * 8;     // N*8   =  0.4M
    unsigned* menc  = (unsigned*)(den + (size_t)N_NODES * 8); // N*8

    const float* hin = x;
    for (int l = 0; l < 2; ++l) {
        const size_t wOff = (size_t)l * 128 * 128;
        const size_t bOff = (size_t)l * 128;
        const size_t eOff = (size_t)l * 4 * 128;

        gemm_qkvs<<<dim3(N_NODES / 16, 4), 256, 0, stream>>>(
            hin, Wq + wOff, bq + bOff, Wk + wOff, bk + bOff,
            Wv + wOff, bv + bOff, Wsk + wOff, bsk + bOff, qkv, agg);

        init_md<<<(N_NODES * 8 + 255) / 256, 256, 0, stream>>>(menc, den, N_NODES * 8);

        edge_logits<<<N_EDGES / 8, 256, 0, stream>>>(qkv, ei, et, Eemb + eOff, exbuf, menc);

        edge_exp<<<(N_EDGES * 8 + 255) / 256, 256, 0, stream>>>(exbuf, ei, menc, den);

        edge_scatter<<<N_EDGES / 8, 256, 0, stream>>>(qkv, ei, et, Eemb + eOff,
                                                      exbuf, den, agg);

        float* out = (l == 0) ? h1 : (float*)d_out;
        ln_relu<<<N_NODES / 8, 256, 0, stream>>>(agg, gamma + bOff, beta + bOff, out);

        hin = h1;
    }
}